// QuantumProjection_59012850647083
// MI455X (gfx1250) — compile-verified
//
#include <hip/hip_runtime.h>
#include <math.h>

typedef __attribute__((ext_vector_type(2))) float v2f;
typedef __attribute__((ext_vector_type(8))) float v8f;

#define BTOT   32768
#define DIN    2048
#define DOUT   2048
#define NQ     4
#define NL     2
#define RPB    64      // rows per block
#define NTHR   256
#define PI_F        3.14159265358979323846f
#define INV_SQRT2_F 0.70710678118654752440f

__global__ __launch_bounds__(NTHR) void qp_fused(
    const float* __restrict__ x,
    const float* __restrict__ Wpre,
    const float* __restrict__ bpre,
    const float* __restrict__ qw,
    const float* __restrict__ Wpost,
    const float* __restrict__ bpost,
    float* __restrict__ out)
{
  __shared__ float sG[NL * NQ * 8];       // 8 rot gates, 8 floats each
  __shared__ float sPart[4][RPB][NQ];     // per-k-quarter partial dots
  __shared__ float sZ[RPB][NQ];           // expectation values

  const int tid      = threadIdx.x;
  const int lane     = tid & 31;
  const int wv       = tid >> 5;
  const int rowGroup = wv & 1;            // which 32-row half of the block
  const int kq       = wv >> 1;           // which 512-wide k quarter
  const int blockRow = blockIdx.x * RPB;

  // ---------- phase 0: batch-constant Rot gate matrices into LDS ----------
  if (tid < NL * NQ) {
    float phi = qw[tid * 3 + 0], th = qw[tid * 3 + 1], om = qw[tid * 3 + 2];
    float s, c;  sincosf(0.5f * th, &s, &c);
    float sp, cp; sincosf(0.5f * (phi + om), &sp, &cp);
    float sm, cm; sincosf(0.5f * (phi - om), &sm, &cm);
    float* g = sG + tid * 8;
    g[0] =  cp * c;  g[1] = -sp * c;    // g00 = e^{-i(phi+om)/2} c
    g[2] = -cm * s;  g[3] = -sm * s;    // g01 = -e^{+i(phi-om)/2} s
    g[4] =  cm * s;  g[5] = -sm * s;    // g10 = e^{-i(phi-om)/2} s
    g[6] =  cp * c;  g[7] =  sp * c;    // g11 = e^{+i(phi+om)/2} c
  }

  // ---------- phase 1: dots = x[row,:] . W_pre[i,:]  (wave-cooperative) ----------
  float4 wpre[4][4];
  {
    const float* wb = Wpre + kq * 512 + lane * 4;
#pragma unroll
    for (int i = 0; i < 4; ++i)
#pragma unroll
      for (int j = 0; j < 4; ++j)
        wpre[i][j] = *(const float4*)(wb + i * DIN + j * 128);
  }

  float pd0 = 0.f, pd1 = 0.f, pd2 = 0.f, pd3 = 0.f;
  for (int r = 0; r < 32; ++r) {
    const float* xr = x + (size_t)(blockRow + rowGroup * 32 + r) * DIN + kq * 512 + lane * 4;
    float a0 = 0.f, a1 = 0.f, a2 = 0.f, a3 = 0.f;
#pragma unroll
    for (int j = 0; j < 4; ++j) {
      float4 xv = *(const float4*)(xr + j * 128);
      a0 += xv.x * wpre[0][j].x + xv.y * wpre[0][j].y + xv.z * wpre[0][j].z + xv.w * wpre[0][j].w;
      a1 += xv.x * wpre[1][j].x + xv.y * wpre[1][j].y + xv.z * wpre[1][j].z + xv.w * wpre[1][j].w;
      a2 += xv.x * wpre[2][j].x + xv.y * wpre[2][j].y + xv.z * wpre[2][j].z + xv.w * wpre[2][j].w;
      a3 += xv.x * wpre[3][j].x + xv.y * wpre[3][j].y + xv.z * wpre[3][j].z + xv.w * wpre[3][j].w;
    }
#pragma unroll
    for (int m = 16; m >= 1; m >>= 1) {
      a0 += __shfl_xor(a0, m);
      a1 += __shfl_xor(a1, m);
      a2 += __shfl_xor(a2, m);
      a3 += __shfl_xor(a3, m);
    }
    if (lane == r) { pd0 = a0; pd1 = a1; pd2 = a2; pd3 = a3; }
  }
  {
    int rr = rowGroup * 32 + lane;
    sPart[kq][rr][0] = pd0; sPart[kq][rr][1] = pd1;
    sPart[kq][rr][2] = pd2; sPart[kq][rr][3] = pd3;
  }
  __syncthreads();

  // ---------- phase 2: 4-qubit circuit, one row per thread (tid < 64) ----------
  if (tid < RPB) {
    float d[4];
#pragma unroll
    for (int i = 0; i < 4; ++i)
      d[i] = sPart[0][tid][i] + sPart[1][tid][i] + sPart[2][tid][i] + sPart[3][tid][i];

    float ar[16], ai[16];
    {
      // per-qubit state after H, RY(atan f), RZ(atan f^2):  v = [e^{-ib/2}(c-s), e^{+ib/2}(c+s)]/sqrt2
      float vr[4][2], vi[4][2];
#pragma unroll
      for (int i = 0; i < 4; ++i) {
        float f  = PI_F * tanhf(d[i] + bpre[i]);
        float al = atanf(f);
        float sa, ca; sincosf(0.5f * al, &sa, &ca);
        float be = atanf(f * f);
        float sb, cb; sincosf(0.5f * be, &sb, &cb);
        float m0 = (ca - sa) * INV_SQRT2_F;
        float m1 = (ca + sa) * INV_SQRT2_F;
        vr[i][0] = cb * m0; vi[i][0] = -sb * m0;
        vr[i][1] = cb * m1; vi[i][1] =  sb * m1;
      }
      // product state: idx s = q0*8 + q1*4 + q2*2 + q3
      float w01r[4], w01i[4], w23r[4], w23i[4];
#pragma unroll
      for (int p = 0; p < 4; ++p) {
        int qa = p >> 1, qb = p & 1;
        w01r[p] = vr[0][qa] * vr[1][qb] - vi[0][qa] * vi[1][qb];
        w01i[p] = vr[0][qa] * vi[1][qb] + vi[0][qa] * vr[1][qb];
        w23r[p] = vr[2][qa] * vr[3][qb] - vi[2][qa] * vi[3][qb];
        w23i[p] = vr[2][qa] * vi[3][qb] + vi[2][qa] * vr[3][qb];
      }
#pragma unroll
      for (int s = 0; s < 16; ++s) {
        int hi = s >> 2, lo = s & 3;
        ar[s] = w01r[hi] * w23r[lo] - w01i[hi] * w23i[lo];
        ai[s] = w01r[hi] * w23i[lo] + w01i[hi] * w23r[lo];
      }
    }

#pragma unroll
    for (int l = 0; l < NL; ++l) {
      // CNOT ring: (control bit, target bit) — qubit i has bit (8 >> i)
      const int cbv[4] = {8, 4, 2, 1};
      const int tbv[4] = {4, 2, 1, 8};
#pragma unroll
      for (int g = 0; g < 4; ++g) {
#pragma unroll
        for (int s = 0; s < 16; ++s) {
          if ((s & cbv[g]) && !(s & tbv[g])) {
            int s2 = s | tbv[g];
            float tr = ar[s]; ar[s] = ar[s2]; ar[s2] = tr;
            float ti = ai[s]; ai[s] = ai[s2]; ai[s2] = ti;
          }
        }
      }
      // Rot gate on each qubit
#pragma unroll
      for (int i = 0; i < 4; ++i) {
        const float* g = sG + (l * 4 + i) * 8;
        float g00r = g[0], g00i = g[1], g01r = g[2], g01i = g[3];
        float g10r = g[4], g10i = g[5], g11r = g[6], g11i = g[7];
        int tb = 8 >> i;
#pragma unroll
        for (int s = 0; s < 16; ++s) {
          if (!(s & tb)) {
            int s1 = s | tb;
            float r0 = ar[s], i0 = ai[s], r1 = ar[s1], i1 = ai[s1];
            ar[s]  = g00r * r0 - g00i * i0 + g01r * r1 - g01i * i1;
            ai[s]  = g00r * i0 + g00i * r0 + g01r * i1 + g01i * r1;
            ar[s1] = g10r * r0 - g10i * i0 + g11r * r1 - g11i * i1;
            ai[s1] = g10r * i0 + g10i * r0 + g11r * i1 + g11i * r1;
          }
        }
      }
    }

    // <Z_i> = sum_s (+/-)|amp_s|^2
#pragma unroll
    for (int i = 0; i < 4; ++i) {
      int tb = 8 >> i;
      float zv = 0.f;
#pragma unroll
      for (int s = 0; s < 16; ++s) {
        float p = ar[s] * ar[s] + ai[s] * ai[s];
        zv += (s & tb) ? -p : p;
      }
      sZ[tid][i] = zv;
    }
  }
  __syncthreads();

  // ---------- phase 3: out = z @ W_post^T + b_post via V_WMMA_F32_16X16X4_F32 ----------
  {
    const int rt = wv & 3;           // row tile (16 rows) within block
    const int ch = wv >> 2;          // column half (64 col-tiles each)
    const int m  = lane & 15;
    const int hi = (lane >> 4) & 1;  // lane half selects K={0,1} vs K={2,3}

    v2f amat;                        // A: 16x4 f32, V0: K=0/2 ; V1: K=1/3
    amat.x = sZ[rt * 16 + m][2 * hi];
    amat.y = sZ[rt * 16 + m][2 * hi + 1];
    const int rowBase = blockRow + rt * 16 + hi * 8;

    for (int t = 0; t < 64; ++t) {
      const int ct  = ch * 64 + t;
      const int col = ct * 16 + m;
      float4 wp = *(const float4*)(Wpost + (size_t)col * 4);  // W_post[col][0..3]
      v2f bmat;                      // B: 4x16 f32, V0: K=0/2 rows ; V1: K=1/3 rows
      bmat.x = hi ? wp.z : wp.x;
      bmat.y = hi ? wp.w : wp.y;
      float bias = bpost[col];
      v8f c;
#pragma unroll
      for (int j = 0; j < 8; ++j) c[j] = bias;   // C[m,n] = b_post[col]
      v8f dacc = __builtin_amdgcn_wmma_f32_16x16x4_f32(
          false, amat, false, bmat, (short)0, c, false, false);
      float* op = out + (size_t)rowBase * DOUT + col;
#pragma unroll
      for (int j = 0; j < 8; ++j) op[(size_t)j * DOUT] = dacc[j];
    }
  }
}

extern "C" void kernel_launch(void* const* d_in, const int* in_sizes, int n_in,
                              void* d_out, int out_size, void* d_ws, size_t ws_size,
                              hipStream_t stream) {
  (void)in_sizes; (void)n_in; (void)d_ws; (void)ws_size; (void)out_size;
  const float* x     = (const float*)d_in[0];
  const float* Wpre  = (const float*)d_in[1];
  const float* bpre  = (const float*)d_in[2];
  const float* qw    = (const float*)d_in[3];
  const float* Wpost = (const float*)d_in[4];
  const float* bpost = (const float*)d_in[5];
  float* out = (float*)d_out;

  qp_fused<<<dim3(BTOT / RPB), NTHR, 0, stream>>>(x, Wpre, bpre, qw, Wpost, bpost, out);
}